// _GraphTransformer_89721866813838
// MI455X (gfx1250) — compile-verified
//
#include <hip/hip_runtime.h>
#include <math.h>

// Problem constants (from reference)
#define BSZ 4
#define NND 160
#define DXC 256
#define DEC 64
#define DYC 64
#define ME_ (BSZ*NND*NND)   // 102400 edge rows
#define MX_ (BSZ*NND)       // 640 node rows
#define LN_EPS 1e-5f
#define INV_SQRT_DF 0.17677669529663687f  // 1/sqrt(32)

typedef __attribute__((ext_vector_type(2)))  float v2f;
typedef __attribute__((ext_vector_type(8)))  float v8f;

// ---------------------------------------------------------------------------
// Common epilogue: mask / relu / store one 16x16 tile
// ---------------------------------------------------------------------------
__device__ __forceinline__ void gt_store_tile(float* __restrict__ OUT, const float* accf,
                                              int mbase, int hi, int ncol, int Nc,
                                              int relu, int accum, int maskmode,
                                              const unsigned char* __restrict__ maskb,
                                              int Nnodes, bool nok) {
#pragma unroll
  for (int r = 0; r < 8; ++r) {
    if (!nok) continue;
    const int m = mbase + r + hi * 8;
    bool mk = true;
    if (maskmode == 1) {
      mk = maskb[m] != 0;
    } else if (maskmode == 2) {
      int j = m % Nnodes; int t = m / Nnodes; int i = t % Nnodes; int b = t / Nnodes;
      mk = (maskb[b * Nnodes + i] != 0) && (maskb[b * Nnodes + j] != 0);
    }
    float v = accf[r];
    if (relu) v = v > 0.f ? v : 0.f;
    if (!mk) { if (accum) continue; v = 0.f; }   // accum: leave prior value intact
    OUT[(size_t)m * Nc + ncol] = v;
  }
}

// ---------------------------------------------------------------------------
// FAST WMMA GEMM: requires K%4==0, Nc%16==0, M%32==0 (true for all hot GEMMs).
// One wave computes two 16x16 M-tiles (rows m0..m0+31), sharing B fragments.
// Branch-free inner loop: A pair -> global_load_b64, B -> 2x global_load_b32.
// ---------------------------------------------------------------------------
__global__ void gt_gemm_wmma_fast(const float* __restrict__ A, const float* __restrict__ W,
                                  const float* __restrict__ bias, float* __restrict__ OUT,
                                  int M, int K, int Nc, int relu, int accum, int maskmode,
                                  const unsigned char* __restrict__ maskb, int Nnodes) {
  const int lane = threadIdx.x & 31;
  const int wave = threadIdx.x >> 5;
  const int m0 = blockIdx.x * 32;
  const int n0 = blockIdx.y * 64 + wave * 16;
  if (n0 >= Nc) return;                 // wave-uniform: EXEC stays all-ones
  const int lh = lane & 15;
  const int hi = lane >> 4;
  const int ncol = n0 + lh;

  union { v8f v; float f[8]; } acc0, acc1;
#pragma unroll
  for (int r = 0; r < 8; ++r) {
    float b0 = bias ? bias[ncol] : 0.f;
    float v0 = b0, v1 = b0;
    if (accum) {
      v0 += OUT[(size_t)(m0 + r + hi * 8) * Nc + ncol];
      v1 += OUT[(size_t)(m0 + 16 + r + hi * 8) * Nc + ncol];
    }
    acc0.f[r] = v0; acc1.f[r] = v1;
  }

  const float* A0 = A + (size_t)(m0 + lh) * K + 2 * hi;   // rows m0..m0+15
  const float* A1 = A0 + (size_t)16 * K;                  // rows m0+16..m0+31
  const float* Wc = W + ncol;

#pragma unroll 4
  for (int k = 0; k < K; k += 4) {
    v2f a0 = *(const v2f*)(A0 + k);                       // A[m][k+2hi], A[m][k+2hi+1]
    v2f a1 = *(const v2f*)(A1 + k);
    union { v2f v; float f[2]; } bf;
    bf.f[0] = Wc[(size_t)(k + 2 * hi) * Nc];              // W[k+2hi][n]
    bf.f[1] = Wc[(size_t)(k + 2 * hi + 1) * Nc];          // W[k+2hi+1][n]
    acc0.v = __builtin_amdgcn_wmma_f32_16x16x4_f32(false, a0, false, bf.v,
                                                   (short)0, acc0.v, false, false);
    acc1.v = __builtin_amdgcn_wmma_f32_16x16x4_f32(false, a1, false, bf.v,
                                                   (short)0, acc1.v, false, false);
  }

  gt_store_tile(OUT, acc0.f, m0,      hi, ncol, Nc, relu, accum, maskmode, maskb, Nnodes, true);
  gt_store_tile(OUT, acc1.f, m0 + 16, hi, ncol, Nc, relu, accum, maskmode, maskb, Nnodes, true);
}

// ---------------------------------------------------------------------------
// GUARDED WMMA GEMM for edge shapes (K=17/5, Nc=17/5). Clamped addresses +
// select-to-zero: unconditional loads, no EXEC-diverging branches in the loop.
// M%16==0 required.
// ---------------------------------------------------------------------------
__global__ void gt_gemm_wmma_guard(const float* __restrict__ A, const float* __restrict__ W,
                                   const float* __restrict__ bias, float* __restrict__ OUT,
                                   int M, int K, int Nc, int relu, int accum, int maskmode,
                                   const unsigned char* __restrict__ maskb, int Nnodes) {
  const int lane = threadIdx.x & 31;
  const int wave = threadIdx.x >> 5;
  const int m0 = blockIdx.x * 16;
  const int n0 = blockIdx.y * 64 + wave * 16;
  if (n0 >= Nc) return;
  const int lh = lane & 15;
  const int hi = lane >> 4;
  const int ncol = n0 + lh;
  const bool nok = (ncol < Nc);
  const int ncc = nok ? ncol : (Nc - 1);                  // clamped column

  union { v8f v; float f[8]; } acc;
#pragma unroll
  for (int r = 0; r < 8; ++r) {
    float v = 0.f;
    if (nok) {
      v = bias ? bias[ncol] : 0.f;
      if (accum) v += OUT[(size_t)(m0 + r + hi * 8) * Nc + ncol];
    }
    acc.f[r] = v;
  }

  const float* Arow = A + (size_t)(m0 + lh) * K;
  for (int k = 0; k < K; k += 4) {
    const int kk = k + 2 * hi;
    const int ka0 = kk     < K ? kk     : K - 1;
    const int ka1 = kk + 1 < K ? kk + 1 : K - 1;
    union { v2f v; float f[2]; } af, bf;
    float a0 = Arow[ka0], a1 = Arow[ka1];
    af.f[0] = (kk     < K) ? a0 : 0.f;
    af.f[1] = (kk + 1 < K) ? a1 : 0.f;
    float b0 = W[(size_t)ka0 * Nc + ncc], b1 = W[(size_t)ka1 * Nc + ncc];
    bf.f[0] = (nok && kk     < K) ? b0 : 0.f;
    bf.f[1] = (nok && kk + 1 < K) ? b1 : 0.f;
    acc.v = __builtin_amdgcn_wmma_f32_16x16x4_f32(false, af.v, false, bf.v,
                                                  (short)0, acc.v, false, false);
  }

  gt_store_tile(OUT, acc.f, m0, hi, ncol, Nc, relu, accum, maskmode, maskb, Nnodes, nok);
}

// Small GEMM for the y-path (M=4): one thread per output element
__global__ void gt_small_lin(const float* __restrict__ A, const float* __restrict__ W,
                             const float* __restrict__ bias, float* __restrict__ OUT,
                             int M, int K, int Nc, int relu, int addTo) {
  int idx = blockIdx.x * blockDim.x + threadIdx.x;
  if (idx >= M * Nc) return;
  int m = idx / Nc, c = idx % Nc;
  float s = bias[c];
  for (int k = 0; k < K; ++k) s += A[(size_t)m * K + k] * W[(size_t)k * Nc + c];
  if (relu) s = s > 0.f ? s : 0.f;
  if (addTo) OUT[idx] += s; else OUT[idx] = s;
}

// out = LayerNorm(a + b) per row, blockDim.x == D (64 or 256)
__global__ void gt_add_ln(const float* __restrict__ a, const float* __restrict__ bsrc,
                          const float* __restrict__ g, const float* __restrict__ beta,
                          float* __restrict__ out, int D) {
  __shared__ float red[256];
  const int row = blockIdx.x, t = threadIdx.x;
  float x = a[(size_t)row * D + t] + bsrc[(size_t)row * D + t];
  red[t] = x; __syncthreads();
  for (int s = D >> 1; s > 0; s >>= 1) { if (t < s) red[t] += red[t + s]; __syncthreads(); }
  float mu = red[0] / (float)D; __syncthreads();
  float d = x - mu;
  red[t] = d * d; __syncthreads();
  for (int s = D >> 1; s > 0; s >>= 1) { if (t < s) red[t] += red[t + s]; __syncthreads(); }
  float var = red[0] / (float)D;
  out[(size_t)row * D + t] = d * rsqrtf(var + LN_EPS) * g[t] + beta[t];
}

// Y = Q*K/sqrt(DF) * (E1 + 1), in place over Ybuf (which holds masked E1)
__global__ void gt_qk_e1(float* __restrict__ Y, const float* __restrict__ Q,
                         const float* __restrict__ Kb, size_t total) {
  size_t idx = (size_t)blockIdx.x * blockDim.x + threadIdx.x;
  if (idx >= total) return;
  int c = (int)(idx & (DXC - 1));
  size_t row = idx >> 8;                 // DXC == 256
  int j  = (int)(row % NND);
  int bi = (int)(row / NND);
  int b  = bi / NND;
  float q = Q[(size_t)bi * DXC + c];
  float k = Kb[((size_t)b * NND + j) * DXC + c];
  Y[idx] = q * k * INV_SQRT_DF * (Y[idx] + 1.f);
}

// buf = add[b] + (mul[b]+1)*buf  (y-conditioning, broadcast over rows)
__global__ void gt_cond(float* __restrict__ buf, const float* __restrict__ addv,
                        const float* __restrict__ mulv, int rowsPerB, int C, size_t total) {
  size_t idx = (size_t)blockIdx.x * blockDim.x + threadIdx.x;
  if (idx >= total) return;
  int c = (int)(idx % C);
  size_t row = idx / C;
  int b = (int)(row / rowsPerB);
  buf[idx] = addv[(size_t)b * C + c] + (mulv[(size_t)b * C + c] + 1.f) * buf[idx];
}

// Per-channel softmax over key axis j, then weighted sum of V -> wV
__global__ void gt_attn(const float* __restrict__ Y, const float* __restrict__ V,
                        const unsigned char* __restrict__ mask, float* __restrict__ out) {
  const int bi = blockIdx.x;      // b*N + i
  const int c  = threadIdx.x;     // channel 0..255
  const int b  = bi / NND;
  float mx = -3.4e38f;
  for (int j = 0; j < NND; ++j)
    if (mask[b * NND + j]) {
      float v = Y[((size_t)bi * NND + j) * DXC + c];
      mx = fmaxf(mx, v);
    }
  float num = 0.f, den = 0.f;
  for (int j = 0; j < NND; ++j)
    if (mask[b * NND + j]) {
      float v = Y[((size_t)bi * NND + j) * DXC + c];
      float e = __expf(v - mx);
      num += e * V[((size_t)b * NND + j) * DXC + c];
      den += e;
    }
  out[(size_t)bi * DXC + c] = (den > 0.f) ? num / den : 0.f;
}

// Eh = (T + T^transpose)/2 * em  (edge symmetrization)
__global__ void gt_sym_e(const float* __restrict__ T, float* __restrict__ out,
                         const unsigned char* __restrict__ mask, size_t total) {
  size_t idx = (size_t)blockIdx.x * blockDim.x + threadIdx.x;
  if (idx >= total) return;
  int c = (int)(idx % DEC);
  size_t r = idx / DEC;
  int j = (int)(r % NND); int t = (int)(r / NND); int i = t % NND; int b = t / NND;
  float v = 0.5f * (T[r * DEC + c] + T[(((size_t)b * NND + j) * NND + i) * DEC + c]);
  bool mk = (mask[b * NND + i] != 0) && (mask[b * NND + j] != 0);
  out[idx] = mk ? v : 0.f;
}

// y-path pooled stats over nodes: zx[b] = concat(mean,min,max,std(ddof=1)) of Xh
__global__ void gt_pool_x(const float* __restrict__ Xh, float* __restrict__ zx) {
  __shared__ float s0[64], s1[64], s2[64], s3[64];
  const int b = blockIdx.x, c = blockIdx.y, t = threadIdx.x;
  float sm = 0.f, sq = 0.f, mn = 3.4e38f, mx = -3.4e38f;
  for (int n = t; n < NND; n += 64) {
    float v = Xh[((size_t)b * NND + n) * DXC + c];
    sm += v; sq += v * v; mn = fminf(mn, v); mx = fmaxf(mx, v);
  }
  s0[t] = sm; s1[t] = sq; s2[t] = mn; s3[t] = mx; __syncthreads();
  for (int s = 32; s > 0; s >>= 1) {
    if (t < s) { s0[t] += s0[t+s]; s1[t] += s1[t+s];
                 s2[t] = fminf(s2[t], s2[t+s]); s3[t] = fmaxf(s3[t], s3[t+s]); }
    __syncthreads();
  }
  if (t == 0) {
    float mean = s0[0] / (float)NND;
    float var  = (s1[0] - (float)NND * mean * mean) / (float)(NND - 1);
    zx[(size_t)b * 4 * DXC + 0 * DXC + c] = mean;
    zx[(size_t)b * 4 * DXC + 1 * DXC + c] = s2[0];
    zx[(size_t)b * 4 * DXC + 2 * DXC + c] = s3[0];
    zx[(size_t)b * 4 * DXC + 3 * DXC + c] = sqrtf(fmaxf(var, 0.f));
  }
}

__global__ void gt_pool_e(const float* __restrict__ Eh, float* __restrict__ ze) {
  __shared__ float s0[128], s1[128], s2[128], s3[128];
  const int b = blockIdx.x, c = blockIdx.y, t = threadIdx.x;
  const int cnt = NND * NND;
  float sm = 0.f, sq = 0.f, mn = 3.4e38f, mx = -3.4e38f;
  for (int n = t; n < cnt; n += 128) {
    float v = Eh[((size_t)b * cnt + n) * DEC + c];
    sm += v; sq += v * v; mn = fminf(mn, v); mx = fmaxf(mx, v);
  }
  s0[t] = sm; s1[t] = sq; s2[t] = mn; s3[t] = mx; __syncthreads();
  for (int s = 64; s > 0; s >>= 1) {
    if (t < s) { s0[t] += s0[t+s]; s1[t] += s1[t+s];
                 s2[t] = fminf(s2[t], s2[t+s]); s3[t] = fmaxf(s3[t], s3[t+s]); }
    __syncthreads();
  }
  if (t == 0) {
    float mean = s0[0] / (float)cnt;
    float var  = (s1[0] - (float)cnt * mean * mean) / (float)(cnt - 1);
    ze[(size_t)b * 4 * DEC + 0 * DEC + c] = mean;
    ze[(size_t)b * 4 * DEC + 1 * DEC + c] = s2[0];
    ze[(size_t)b * 4 * DEC + 2 * DEC + c] = s3[0];
    ze[(size_t)b * 4 * DEC + 3 * DEC + c] = sqrtf(fmaxf(var, 0.f));
  }
}

// Final outputs
__global__ void gt_final_x(const float* __restrict__ G, const float* __restrict__ X0,
                           const unsigned char* __restrict__ mask, float* __restrict__ out) {
  int idx = blockIdx.x * blockDim.x + threadIdx.x;
  if (idx >= MX_ * 17) return;
  int r = idx / 17;
  out[idx] = (G[idx] + X0[idx]) * (mask[r] ? 1.f : 0.f);
}

__global__ void gt_final_e(const float* __restrict__ G, const float* __restrict__ E0,
                           const unsigned char* __restrict__ mask, float* __restrict__ out) {
  size_t idx = (size_t)blockIdx.x * blockDim.x + threadIdx.x;
  if (idx >= (size_t)ME_ * 5) return;
  int c = (int)(idx % 5);
  size_t r = idx / 5;
  int j = (int)(r % NND); int t = (int)(r / NND); int i = t % NND; int b = t / NND;
  float g1 = G[r * 5 + c];
  float g2 = G[(((size_t)b * NND + j) * NND + i) * 5 + c];
  float d = (i != j) ? 1.f : 0.f;
  bool mk = (mask[b * NND + i] != 0) && (mask[b * NND + j] != 0);
  out[idx] = mk ? (0.5f * (g1 + g2) * d + E0[r * 5 + c]) : 0.f;
}

__global__ void gt_final_y(const float* __restrict__ G, const float* __restrict__ y0,
                           float* __restrict__ out) {
  int idx = blockIdx.x * blockDim.x + threadIdx.x;
  if (idx >= BSZ * 13) return;
  out[idx] = G[idx] + y0[idx];
}

// ---------------------------------------------------------------------------
extern "C" void kernel_launch(void* const* d_in, const int* in_sizes, int n_in,
                              void* d_out, int out_size, void* d_ws, size_t ws_size,
                              hipStream_t stream) {
  const float* Xin = (const float*)d_in[0];
  const float* Ein = (const float*)d_in[1];
  const float* yin = (const float*)d_in[2];
  const unsigned char* maskb = (const unsigned char*)d_in[3];

  // Param leaves, assumed jax-pytree (sorted-key) flatten order of `params`
  const float* P[136];
  for (int i = 0; i < 136; ++i) P[i] = (4 + i < n_in) ? (const float*)d_in[4 + i] : nullptr;
  // params: in_E(0..3) in_X(4..7) in_y(8..11) layers(12..123, 56/layer)
  //         out_E(124..127) out_X(128..131) out_y(132..135)
  // layer:  e_add(b0 w1) e_mul(2 3) e_out(4 5) e_y(6 7) k(8 9) q(10 11) v(12 13)
  //         x_out(14 15) x_y(16 17) y_e_add(18 19) y_e_mul(20 21) y_out1(22 23)
  //         y_out2(24 25) y_x_add(26 27) y_x_mul(28 29) y_y(30 31)
  //         linE1(32 33) linE2(34 35) linX1(36 37) linX2(38 39) lin_y1(40 41) lin_y2(42 43)
  //         normE1(b44 g45) normE2(46 47) normX1(48 49) normX2(50 51) norm_y1(52 53) norm_y2(54 55)

  float* Wf = (float*)d_ws;
  size_t off = 0;
  float* Eh    = Wf + off; off += (size_t)ME_ * DEC;   // 6.55M
  float* Enew  = Wf + off; off += (size_t)ME_ * DEC;   // 6.55M
  float* Ybuf  = Wf + off; off += (size_t)ME_ * DXC;   // 26.2M (also E-MLP hidden)
  float* Xh    = Wf + off; off += (size_t)MX_ * DXC;
  float* Xnew  = Wf + off; off += (size_t)MX_ * DXC;
  float* Xtmp  = Wf + off; off += (size_t)MX_ * DXC;
  float* Xattn = Wf + off; off += (size_t)MX_ * DXC;
  float* Qb    = Wf + off; off += (size_t)MX_ * DXC;
  float* Kb    = Wf + off; off += (size_t)MX_ * DXC;
  float* Vb    = Wf + off; off += (size_t)MX_ * DXC;
  float* yh    = Wf + off; off += 512;
  float* YB1   = Wf + off; off += 512;
  float* YB2   = Wf + off; off += 512;
  float* YB3   = Wf + off; off += 512;
  float* YEA   = Wf + off; off += 1024;
  float* YEM   = Wf + off; off += 1024;
  float* YXA   = Wf + off; off += 1024;
  float* YXM   = Wf + off; off += 1024;
  float* ZX    = Wf + off; off += 4096;
  float* ZE    = Wf + off; off += 1024;

  float* outX = (float*)d_out;
  float* outE = outX + MX_ * 17;
  float* outY = outE + (size_t)ME_ * 5;

  auto GEMM = [&](const float* A, const float* Wt, const float* bias, float* OUT,
                  int M, int K, int Nc, int relu, int accum, int maskmode) {
    const bool guard = (K % 4 != 0) || (Nc % 16 != 0) || (M % 32 != 0);
    if (!guard) {
      dim3 g(M / 32, (Nc + 63) / 64);
      gt_gemm_wmma_fast<<<g, 128, 0, stream>>>(A, Wt, bias, OUT, M, K, Nc,
                                               relu, accum, maskmode, maskb, NND);
    } else {
      dim3 g(M / 16, (Nc + 63) / 64);
      gt_gemm_wmma_guard<<<g, 128, 0, stream>>>(A, Wt, bias, OUT, M, K, Nc,
                                                relu, accum, maskmode, maskb, NND);
    }
  };
  auto SLIN = [&](const float* A, const float* Wt, const float* bias, float* OUT,
                  int M, int K, int Nc, int relu, int addTo) {
    int tot = M * Nc;
    gt_small_lin<<<(tot + 255) / 256, 256, 0, stream>>>(A, Wt, bias, OUT, M, K, Nc, relu, addTo);
  };
  auto ADDLN = [&](const float* a, const float* b, const float* g, const float* be,
                   float* out, int rows, int D) {
    gt_add_ln<<<rows, D, 0, stream>>>(a, b, g, be, out, D);
  };

  // ---- Input embeddings ----
  GEMM(Xin, P[5], P[4], Xtmp, MX_, 17, 256, 1, 0, 0);          // in_X L0 + relu (guarded K)
  GEMM(Xtmp, P[7], P[6], Xh,  MX_, 256, DXC, 1, 0, 1);         // in_X L1 + relu, *xm
  GEMM(Ein, P[1], P[0], Ybuf, ME_, 5, 128, 1, 0, 0);           // in_E L0 + relu (guarded K)
  GEMM(Ybuf, P[3], P[2], Enew, ME_, 128, DEC, 1, 0, 0);        // in_E L1 + relu
  gt_sym_e<<<(ME_ * DEC + 255) / 256, 256, 0, stream>>>(Enew, Eh, maskb, (size_t)ME_ * DEC);
  SLIN(yin, P[9], P[8], YB2, BSZ, 13, 128, 1, 0);              // in_y L0 + relu
  SLIN(YB2, P[11], P[10], yh, BSZ, 128, DYC, 1, 0);            // in_y L1 + relu

  // ---- Transformer layers ----
  for (int layer = 0; layer < 2; ++layer) {
    const int lb = 12 + 56 * layer;
    // y-conditioning vectors
    SLIN(yh, P[lb + 19], P[lb + 18], YEA, BSZ, DYC, DXC, 0, 0);   // y_e_add
    SLIN(yh, P[lb + 21], P[lb + 20], YEM, BSZ, DYC, DXC, 0, 0);   // y_e_mul
    SLIN(yh, P[lb + 27], P[lb + 26], YXA, BSZ, DYC, DXC, 0, 0);   // y_x_add
    SLIN(yh, P[lb + 29], P[lb + 28], YXM, BSZ, DYC, DXC, 0, 0);   // y_x_mul
    // Q,K,V (masked)
    GEMM(Xh, P[lb + 11], P[lb + 10], Qb, MX_, DXC, DXC, 0, 0, 1);
    GEMM(Xh, P[lb +  9], P[lb +  8], Kb, MX_, DXC, DXC, 0, 0, 1);
    GEMM(Xh, P[lb + 13], P[lb + 12], Vb, MX_, DXC, DXC, 0, 0, 1);
    // Y = QK/s * (E1+1) + E2 built in Ybuf
    GEMM(Eh, P[lb + 3], P[lb + 2], Ybuf, ME_, DEC, DXC, 0, 0, 2);     // E1 (masked)
    {
      size_t tot = (size_t)ME_ * DXC;
      gt_qk_e1<<<(unsigned)((tot + 255) / 256), 256, 0, stream>>>(Ybuf, Qb, Kb, tot);
    }
    GEMM(Eh, P[lb + 1], P[lb + 0], Ybuf, ME_, DEC, DXC, 0, 1, 2);     // += E2 (masked, accum)
    // Attention (per-channel softmax over keys) -> wV
    gt_attn<<<MX_, DXC, 0, stream>>>(Ybuf, Vb, maskb, Xattn);
    // newE: condition Y in place, project with e_out
    {
      size_t tot = (size_t)ME_ * DXC;
      gt_cond<<<(unsigned)((tot + 255) / 256), 256, 0, stream>>>(Ybuf, YEA, YEM, NND * NND, DXC, tot);
    }
    GEMM(Ybuf, P[lb + 5], P[lb + 4], Enew, ME_, DXC, DEC, 0, 0, 2);   // e_out (masked)
    // newX: condition wV, project with x_out
    {
      size_t tot = (size_t)MX_ * DXC;
      gt_cond<<<(unsigned)((tot + 255) / 256), 256, 0, stream>>>(Xattn, YXA, YXM, NND, DXC, tot);
    }
    GEMM(Xattn, P[lb + 15], P[lb + 14], Xnew, MX_, DXC, DXC, 0, 0, 1); // x_out (masked)
    // new_y = y_y(y) + x_y(pool(X)) + e_y(pool(E)), then 2-layer MLP
    gt_pool_x<<<dim3(BSZ, DXC), 64, 0, stream>>>(Xh, ZX);
    gt_pool_e<<<dim3(BSZ, DEC), 128, 0, stream>>>(Eh, ZE);
    SLIN(yh, P[lb + 31], P[lb + 30], YB1, BSZ, DYC, DYC, 0, 0);       // y_y
    SLIN(ZX, P[lb + 17], P[lb + 16], YB1, BSZ, 4 * DXC, DYC, 0, 1);   // + x_y
    SLIN(ZE, P[lb +  7], P[lb +  6], YB1, BSZ, 4 * DEC, DYC, 0, 1);   // + e_y
    SLIN(YB1, P[lb + 23], P[lb + 22], YB2, BSZ, DYC, DYC, 1, 0);      // y_out1 + relu
    SLIN(YB2, P[lb + 25], P[lb + 24], YB3, BSZ, DYC, DYC, 0, 0);      // y_out2
    // Residual + LN (block 1)
    ADDLN(Xh, Xnew, P[lb + 49], P[lb + 48], Xh, MX_, DXC);            // normX1
    ADDLN(Eh, Enew, P[lb + 45], P[lb + 44], Eh, ME_, DEC);            // normE1
    ADDLN(yh, YB3,  P[lb + 53], P[lb + 52], yh, BSZ, DYC);            // norm_y1
    // FFNs + LN (block 2)
    GEMM(Xh, P[lb + 37], P[lb + 36], Xtmp, MX_, DXC, 256, 1, 0, 0);   // linX1 + relu
    GEMM(Xtmp, P[lb + 39], P[lb + 38], Xnew, MX_, 256, DXC, 0, 0, 0); // linX2
    ADDLN(Xh, Xnew, P[lb + 51], P[lb + 50], Xh, MX_, DXC);            // normX2
    GEMM(Eh, P[lb + 33], P[lb + 32], Ybuf, ME_, DEC, 128, 1, 0, 0);   // linE1 + relu
    GEMM(Ybuf, P[lb + 35], P[lb + 34], Enew, ME_, 128, DEC, 0, 0, 0); // linE2
    ADDLN(Eh, Enew, P[lb + 47], P[lb + 46], Eh, ME_, DEC);            // normE2
    SLIN(yh, P[lb + 41], P[lb + 40], YB2, BSZ, DYC, 128, 1, 0);       // lin_y1 + relu
    SLIN(YB2, P[lb + 43], P[lb + 42], YB3, BSZ, 128, DYC, 0, 0);      // lin_y2
    ADDLN(yh, YB3, P[lb + 55], P[lb + 54], yh, BSZ, DYC);             // norm_y2
  }

  // ---- Output heads ----
  GEMM(Xh, P[129], P[128], Xtmp, MX_, DXC, 256, 1, 0, 0);       // out_X L0 + relu
  GEMM(Xtmp, P[131], P[130], Xnew, MX_, 256, 17, 0, 0, 0);      // out_X L1 (guarded Nc)
  gt_final_x<<<(MX_ * 17 + 255) / 256, 256, 0, stream>>>(Xnew, Xin, maskb, outX);

  GEMM(Eh, P[125], P[124], Ybuf, ME_, DEC, 128, 1, 0, 0);       // out_E L0 + relu
  GEMM(Ybuf, P[127], P[126], Enew, ME_, 128, 5, 0, 0, 0);       // out_E L1 (guarded Nc)
  {
    size_t tot = (size_t)ME_ * 5;
    gt_final_e<<<(unsigned)((tot + 255) / 256), 256, 0, stream>>>(Enew, Ein, maskb, outE);
  }

  SLIN(yh, P[133], P[132], YB2, BSZ, DYC, 128, 1, 0);           // out_y L0 + relu
  SLIN(YB2, P[135], P[134], YB3, BSZ, 128, 13, 0, 0);           // out_y L1
  gt_final_y<<<1, 64, 0, stream>>>(YB3, yin, outY);
}